// path_16269336117656
// MI455X (gfx1250) — compile-verified
//
#include <hip/hip_runtime.h>

// ---------------------------------------------------------------------------
// Physics-integration scan for MI455X (gfx1250, wave32).
//
//   vel_{t+1} = vel0 + dt*S_t,   S_t = sum_{i<=t} a_i
//   pos_{t+1} = pos0 + (t+1)*dt*vel0 + dt^2*((t+0.5)*S_t - T_t),
//               T_t = sum_{i<=t} i*a_i
//
// 3-phase deterministic parallel scan (no atomics, graph-capture safe):
//   A) per-(channel,chunk) sums of a and t*a          -> ws[0..383]
//   B) tiny exclusive scan over 32 chunks per channel -> ws[384..767]
//   C) intra-chunk inclusive cumsum via V_WMMA_F32_16X16X4_F32
//      (cumsum = L * D, L = lower-triangular ones, chained over K),
//      then emit pos/vel and the actions copy.
// ---------------------------------------------------------------------------

typedef __attribute__((ext_vector_type(2))) float v2f;
typedef __attribute__((ext_vector_type(8))) float v8f;

#define STEPS   65536
#define NCH     6
#define CHUNK   2048
#define NCHUNK  (STEPS / CHUNK)   // 32
#define DT      0.1f
#define TILEPAD 272               // 16x17 padded tile (bank-conflict-free cols)

// ---- Kernel A: chunk partial sums (S and T) -------------------------------
__global__ void partials_kernel(const float* __restrict__ actions,
                                float* __restrict__ ws) {
    const int cx = blockIdx.x, c = blockIdx.y, tid = threadIdx.x;
    const int t0 = cx * CHUNK;

    float s = 0.f, tt = 0.f;
    for (int k = 0; k < CHUNK / 256; ++k) {
        const int t = t0 + tid + k * 256;
        const float a = actions[t * NCH + c];
        s  += a;
        tt += (float)t * a;
    }

    __shared__ float rs[256], rt[256];
    rs[tid] = s; rt[tid] = tt;
    __syncthreads();
    for (int off = 128; off > 0; off >>= 1) {
        if (tid < off) { rs[tid] += rs[tid + off]; rt[tid] += rt[tid + off]; }
        __syncthreads();
    }
    if (tid == 0) {
        ws[(c * NCHUNK + cx) * 2 + 0] = rs[0];
        ws[(c * NCHUNK + cx) * 2 + 1] = rt[0];
    }
}

// ---- Kernel B: exclusive scan over the 32 chunk partials (12 tiny scans) ---
__global__ void chunk_scan_kernel(float* __restrict__ ws) {
    const int tid = threadIdx.x;
    if (tid < NCH * 2) {
        const int c = tid >> 1, q = tid & 1;
        float run = 0.f;
        for (int cx = 0; cx < NCHUNK; ++cx) {
            const int idx = (c * NCHUNK + cx) * 2 + q;
            ws[NCH * NCHUNK * 2 + idx] = run;   // exclusive prefix
            run += ws[idx];
        }
    }
}

// ---- Kernel C: WMMA intra-chunk cumsum + output ---------------------------
// Block = 256 threads = 8 waves; each wave owns one 256-element 16x16 tile,
// column-major (local flat f = n*16 + m), padded stride 17 in LDS.
__global__ void finalize_kernel(const float* __restrict__ x,
                                const float* __restrict__ actions,
                                const float* __restrict__ ws,
                                float* __restrict__ out) {
    const int cx  = blockIdx.x, c = blockIdx.y;
    const int tid = threadIdx.x;
    const int w   = tid >> 5;          // wave in block (0..7)
    const int l   = tid & 31;          // lane (wave32)
    const int t0  = cx * CHUNK;

    __shared__ float sData[8 * TILEPAD];
    __shared__ float sWS[8], sWT[8];

    // Stage channel data: global stride-6 gather -> padded column-major tiles.
    for (int j = tid; j < CHUNK; j += 256) {
        const int tile = j >> 8, f = j & 255;
        sData[tile * TILEPAD + (f >> 4) * 17 + (f & 15)] =
            actions[(t0 + j) * NCH + c];
    }
    __syncthreads();

    const int   tb     = w * TILEPAD;
    const int   n      = l & 15;       // column (also A-row m for this lane)
    const int   half   = l >> 4;       // 0: lanes 0-15, 1: lanes 16-31
    const int   tileT0 = t0 + w * 256; // global t of local flat index 0

    // Inclusive column cumsum: acc = L16x16 * D, chained over K in 4 slices.
    // A layout (f32 16x4): lane -> M=l%16; VGPR0/1 hold K = 2*half + {0,1}.
    // B layout (f32 4x16): lane -> N=l%16; VGPR0/1 hold K = 2*half + {0,1}.
    v8f accS = {};
    v8f accT = {};
    for (int j = 0; j < 4; ++j) {
        const int kbase = 4 * j + 2 * half;
        v2f a, bS, bT;
        a.x = (kbase     <= n) ? 1.f : 0.f;   // L[m][k]  (m == l%16)
        a.y = (kbase + 1 <= n) ? 1.f : 0.f;
        const float d0 = sData[tb + n * 17 + kbase];
        const float d1 = sData[tb + n * 17 + kbase + 1];
        bS.x = d0; bS.y = d1;
        const float tg = (float)(tileT0 + n * 16 + kbase);
        bT.x = tg * d0; bT.y = (tg + 1.f) * d1;
        accS = __builtin_amdgcn_wmma_f32_16x16x4_f32(
                   false, a, false, bS, (short)0, accS, false, false);
        accT = __builtin_amdgcn_wmma_f32_16x16x4_f32(
                   false, a, false, bT, (short)0, accT, false, false);
    }

    // Cross-column offsets: column sums live in VGPR7, lanes 16..31 (M=15).
    float offS = 0.f, offT = 0.f, totS = 0.f, totT = 0.f;
    const float c7s = accS[7], c7t = accT[7];
    for (int i = 0; i < 16; ++i) {
        const float vs = __shfl(c7s, 16 + i, 32);
        const float vt = __shfl(c7t, 16 + i, 32);
        if (i < n) { offS += vs; offT += vt; }
        totS += vs; totT += vt;
    }

    // Cross-wave offsets within the chunk.
    if (l == 0) { sWS[w] = totS; sWT[w] = totT; }
    __syncthreads();
    float wOffS = 0.f, wOffT = 0.f;
    for (int i = 0; i < 8; ++i) {
        if (i < w) { wOffS += sWS[i]; wOffT += sWT[i]; }
    }

    // Cross-chunk exclusive offsets from phase B.
    const float exS = ws[NCH * NCHUNK * 2 + (c * NCHUNK + cx) * 2 + 0];
    const float exT = ws[NCH * NCHUNK * 2 + (c * NCHUNK + cx) * 2 + 1];

    const float pos0 = x[c], vel0 = x[6 + c];
    const float dt = DT, dt2 = DT * DT;
    const float baseS = offS + wOffS + exS;
    const float baseT = offT + wOffT + exT;

    float* __restrict__ statesOut = out;               // [STEPS][12]
    float* __restrict__ actOut    = out + STEPS * 12;  // [STEPS][6]

    // C/D layout (f32 16x16): VGPR v, lane l -> M = v + 8*half, N = l%16.
    for (int v = 0; v < 8; ++v) {
        const int m = v + 8 * half;
        const int f = n * 16 + m;
        const int t = tileT0 + f;
        const float S  = accS[v] + baseS;
        const float T  = accT[v] + baseT;
        const float tf = (float)t;
        const float vel = vel0 + dt * S;
        const float pos = pos0 + (tf + 1.f) * dt * vel0
                               + dt2 * ((tf + 0.5f) * S - T);
        statesOut[t * 12 + c]     = pos;
        statesOut[t * 12 + 6 + c] = vel;
        actOut[t * NCH + c]       = sData[tb + n * 17 + m];
    }
}

extern "C" void kernel_launch(void* const* d_in, const int* in_sizes, int n_in,
                              void* d_out, int out_size, void* d_ws, size_t ws_size,
                              hipStream_t stream) {
    const float* x       = (const float*)d_in[0];   // [12] = pos(6), vel(6)
    const float* actions = (const float*)d_in[1];   // [65536, 6]
    float*       out     = (float*)d_out;           // [65536*12] + [65536*6]
    float*       ws      = (float*)d_ws;            // 768 floats used

    partials_kernel  <<<dim3(NCHUNK, NCH), 256, 0, stream>>>(actions, ws);
    chunk_scan_kernel<<<1, 32, 0, stream>>>(ws);
    finalize_kernel  <<<dim3(NCHUNK, NCH), 256, 0, stream>>>(x, actions, ws, out);
}